// SSILoss_55104430408338
// MI455X (gfx1250) — compile-verified
//
#include <hip/hip_runtime.h>
#include <hip/hip_bf16.h>

// ---------------------------------------------------------------------------
// SSI loss for MI455X (gfx1250, wave32).
//   B=32 batches, N=307200 f32 elements/batch. Masked median via 3-pass
//   radix-select (11/11/10-bit digits, 2048-bin LDS histograms, deterministic
//   integer atomics) on the order-preserving uint32 mapping of f32.
//   All streaming loops use 128-bit loads (float4/uchar4). Final f32
//   reductions use V_WMMA_F32_16X16X4_F32 (ones-matrix: exact f32 multiply-
//   by-1 + f32 accumulate, fixed order => deterministic).
//   Data set (~94 MB) is L2-resident on the 192 MB L2; 5 element passes total.
// ---------------------------------------------------------------------------

#define BATCHES 32
#define NB      64      // chunks (blocks) per batch for element-scan kernels
#define TPB     256
#define HBINS   2048    // max radix bins (pass 0/1: 2048, pass 2: 1024)

// Workspace layout in 32-bit words (~550 KB total).
enum : int {
  WS_NVALID  = 0,                           // u32[32]
  WS_TMINU   = 32,                          // u32[32] mapped-uint min of masked target
  WS_TMAXU   = 64,                          // u32[32] mapped-uint max of masked target
  WS_PREF_I  = 96,                          // u32[32] radix-select prefix (input)
  WS_KREM_I  = 128,                         // u32[32] remaining k (input)
  WS_PREF_T  = 160,                         // u32[32] prefix (target)
  WS_KREM_T  = 192,                         // u32[32] remaining k (target)
  WS_MED_I   = 224,                         // f32[32]
  WS_MED_T   = 256,                         // f32[32]
  WS_SCALE_I = 288,                         // f32[32]
  WS_SCALE_T = 320,                         // f32[32]
  WS_HIST_I  = 384,                         // u32[32][2048]
  WS_HIST_T  = WS_HIST_I + BATCHES * HBINS, // u32[32][2048]
  WS_PART_SI = WS_HIST_T + BATCHES * HBINS, // f32[32*64]
  WS_PART_ST = WS_PART_SI + BATCHES * NB,   // f32[32*64]
  WS_PART_L  = WS_PART_ST + BATCHES * NB,   // f32[32*64]
  WS_TOTAL   = WS_PART_L + BATCHES * NB
};

typedef __attribute__((ext_vector_type(2))) float v2f;
typedef __attribute__((ext_vector_type(8))) float v8f;

// Order-preserving f32 -> u32 mapping (ascending).
__device__ __forceinline__ unsigned fmap(float f) {
  unsigned u = __float_as_uint(f);
  return (u & 0x80000000u) ? ~u : (u ^ 0x80000000u);
}
__device__ __forceinline__ float funmap(unsigned u) {
  unsigned b = (u & 0x80000000u) ? (u ^ 0x80000000u) : ~u;
  return __uint_as_float(b);
}

// Deterministic fixed-order block sum (256 threads).
__device__ __forceinline__ float block_sum_f(float v, float* s) {
  int tid = threadIdx.x;
  s[tid] = v; __syncthreads();
  #pragma unroll
  for (int w = TPB / 2; w > 0; w >>= 1) {
    if (tid < w) s[tid] += s[tid + w];
    __syncthreads();
  }
  float r = s[0]; __syncthreads();
  return r;
}

// Exact f32 wave reduction of `count` (multiple of 64) floats using
// V_WMMA_F32_16X16X4_F32: D = ones(16x4) * B(4x16) + C => D[m][n] = column
// sums of B; each input lands in exactly one B slot, so the sum over one row
// of D equals the total. Accumulates chunks through C. Lane 0 returns total.
__device__ __forceinline__ float wave_sum_mult64(const float* p, int count) {
  int lane = threadIdx.x & 31;
#if __has_builtin(__builtin_amdgcn_wmma_f32_16x16x4_f32)
  v2f a; a[0] = 1.0f; a[1] = 1.0f;   // A = all ones (16x4 striped over wave)
  v8f c = {};
  for (int base = 0; base < count; base += 64) {
    v2f bv; bv[0] = p[base + lane * 2]; bv[1] = p[base + lane * 2 + 1];
    c = __builtin_amdgcn_wmma_f32_16x16x4_f32(
        /*neg_a=*/false, a, /*neg_b=*/false, bv,
        /*c_mod=*/(short)0, c, /*reuse_a=*/false, /*reuse_b=*/false);
  }
  float s = c[0];                 // row M=0: lane n<16 holds column-sum n
  s += __shfl_xor(s, 1, 32);
  s += __shfl_xor(s, 2, 32);
  s += __shfl_xor(s, 4, 32);
  s += __shfl_xor(s, 8, 32);      // lanes 0..15 group -> total in lane 0
  return s;
#else
  float s = 0.0f;
  for (int i = lane; i < count; i += 32) s += p[i];
  #pragma unroll
  for (int d = 1; d < 32; d <<= 1) s += __shfl_xor(s, d, 32);
  return s;
#endif
}

// ---------------------------------------------------------------------------
__global__ void ssi_init(unsigned* __restrict__ ws) {
  for (int i = blockIdx.x * TPB + threadIdx.x; i < WS_TOTAL; i += gridDim.x * TPB) {
    unsigned v = 0u;
    if (i >= WS_TMINU && i < WS_TMINU + BATCHES) v = 0xFFFFFFFFu;
    ws[i] = v;
  }
}

// --- per-element helpers ----------------------------------------------------
__device__ __forceinline__ void stat_elem(float fi, float ft, bool m,
                                          unsigned* shi, unsigned* sht,
                                          unsigned& cnt, unsigned& mn, unsigned& mx) {
  unsigned ui = m ? fmap(fi) : 0xFFFFFFFFu;   // invalid -> ordered last
  unsigned ut = m ? fmap(ft) : 0xFFFFFFFFu;
  if (m) { cnt++; mn = (ut < mn) ? ut : mn; mx = (ut > mx) ? ut : mx; }
  atomicAdd(&shi[ui >> 21], 1u);              // pass-0 digit: bits [31:21]
  atomicAdd(&sht[ut >> 21], 1u);
}

__device__ __forceinline__ void hist_elem(float fi, float ft, bool m, int pass,
                                          unsigned prefi, unsigned preft,
                                          unsigned* shi, unsigned* sht) {
  unsigned ui = m ? fmap(fi) : 0xFFFFFFFFu;
  unsigned ut = m ? fmap(ft) : 0xFFFFFFFFu;
  if (pass == 1) {                            // digit bits [20:10]
    if ((ui >> 21) == prefi) atomicAdd(&shi[(ui >> 10) & 0x7FFu], 1u);
    if ((ut >> 21) == preft) atomicAdd(&sht[(ut >> 10) & 0x7FFu], 1u);
  } else {                                    // pass 2: digit bits [9:0]
    if ((ui >> 10) == prefi) atomicAdd(&shi[ui & 0x3FFu], 1u);
    if ((ut >> 10) == preft) atomicAdd(&sht[ut & 0x3FFu], 1u);
  }
}

// Pass 0: nvalid, masked target min/max (mapped uints), 2048-bin histograms of
// the top 11 bits of mapped input & target.
__global__ void ssi_stats(const float* __restrict__ xi, const float* __restrict__ xt,
                          const unsigned char* __restrict__ mask,
                          unsigned* __restrict__ ws, int n) {
  __shared__ unsigned shi[HBINS], sht[HBINS], sred[TPB];
  int tid = threadIdx.x;
  for (int i = tid; i < HBINS; i += TPB) { shi[i] = 0u; sht[i] = 0u; }
  __syncthreads();

  int b = blockIdx.x >> 6, chunk = blockIdx.x & (NB - 1);
  const float4* pi4 = (const float4*)(xi + (size_t)b * n);
  const float4* pt4 = (const float4*)(xt + (size_t)b * n);
  const uchar4* pm4 = (const uchar4*)(mask + (size_t)b * n);
  int nv4 = n >> 2;

  unsigned cnt = 0u, mn = 0xFFFFFFFFu, mx = 0u;
  for (int v = chunk * TPB + tid; v < nv4; v += NB * TPB) {
    float4 vi = pi4[v]; float4 vt = pt4[v]; uchar4 m4 = pm4[v];
    stat_elem(vi.x, vt.x, m4.x != 0, shi, sht, cnt, mn, mx);
    stat_elem(vi.y, vt.y, m4.y != 0, shi, sht, cnt, mn, mx);
    stat_elem(vi.z, vt.z, m4.z != 0, shi, sht, cnt, mn, mx);
    stat_elem(vi.w, vt.w, m4.w != 0, shi, sht, cnt, mn, mx);
  }
  if (chunk == 0) {  // scalar tail (n % 4 != 0); no-op for this problem
    const float* pi = xi + (size_t)b * n;
    const float* pt = xt + (size_t)b * n;
    const unsigned char* pm = mask + (size_t)b * n;
    for (int e = (nv4 << 2) + tid; e < n; e += TPB)
      stat_elem(pi[e], pt[e], pm[e] != 0, shi, sht, cnt, mn, mx);
  }
  __syncthreads();
  for (int i = tid; i < HBINS; i += TPB) {
    atomicAdd(&ws[WS_HIST_I + b * HBINS + i], shi[i]);
    atomicAdd(&ws[WS_HIST_T + b * HBINS + i], sht[i]);
  }

  // block reductions (deterministic; cross-block via integer atomics)
  sred[tid] = cnt; __syncthreads();
  for (int w = TPB / 2; w > 0; w >>= 1) { if (tid < w) sred[tid] += sred[tid + w]; __syncthreads(); }
  if (tid == 0) atomicAdd(&ws[WS_NVALID + b], sred[0]);
  __syncthreads();
  sred[tid] = mn; __syncthreads();
  for (int w = TPB / 2; w > 0; w >>= 1) { if (tid < w && sred[tid + w] < sred[tid]) sred[tid] = sred[tid + w]; __syncthreads(); }
  if (tid == 0) atomicMin(&ws[WS_TMINU + b], sred[0]);
  __syncthreads();
  sred[tid] = mx; __syncthreads();
  for (int w = TPB / 2; w > 0; w >>= 1) { if (tid < w && sred[tid + w] > sred[tid]) sred[tid] = sred[tid + w]; __syncthreads(); }
  if (tid == 0) atomicMax(&ws[WS_TMAXU + b], sred[0]);
}

// Histogram passes 1..2: bin next digit of elements whose prefix matches.
__global__ void ssi_hist(const float* __restrict__ xi, const float* __restrict__ xt,
                         const unsigned char* __restrict__ mask,
                         unsigned* __restrict__ ws, int n, int pass) {
  __shared__ unsigned shi[HBINS], sht[HBINS];
  int tid = threadIdx.x;
  for (int i = tid; i < HBINS; i += TPB) { shi[i] = 0u; sht[i] = 0u; }
  __syncthreads();

  int b = blockIdx.x >> 6, chunk = blockIdx.x & (NB - 1);
  int hishift = (pass == 1) ? 21 : 10;
  unsigned prefi = ws[WS_PREF_I + b] >> hishift;
  unsigned preft = ws[WS_PREF_T + b] >> hishift;

  const float4* pi4 = (const float4*)(xi + (size_t)b * n);
  const float4* pt4 = (const float4*)(xt + (size_t)b * n);
  const uchar4* pm4 = (const uchar4*)(mask + (size_t)b * n);
  int nv4 = n >> 2;

  for (int v = chunk * TPB + tid; v < nv4; v += NB * TPB) {
    float4 vi = pi4[v]; float4 vt = pt4[v]; uchar4 m4 = pm4[v];
    hist_elem(vi.x, vt.x, m4.x != 0, pass, prefi, preft, shi, sht);
    hist_elem(vi.y, vt.y, m4.y != 0, pass, prefi, preft, shi, sht);
    hist_elem(vi.z, vt.z, m4.z != 0, pass, prefi, preft, shi, sht);
    hist_elem(vi.w, vt.w, m4.w != 0, pass, prefi, preft, shi, sht);
  }
  if (chunk == 0) {
    const float* pi = xi + (size_t)b * n;
    const float* pt = xt + (size_t)b * n;
    const unsigned char* pm = mask + (size_t)b * n;
    for (int e = (nv4 << 2) + tid; e < n; e += TPB)
      hist_elem(pi[e], pt[e], pm[e] != 0, pass, prefi, preft, shi, sht);
  }
  __syncthreads();
  for (int i = tid; i < HBINS; i += TPB) {
    atomicAdd(&ws[WS_HIST_I + b * HBINS + i], shi[i]);
    atomicAdd(&ws[WS_HIST_T + b * HBINS + i], sht[i]);
  }
}

// Parallel digit selection: per-thread partial sums + block prefix scan to
// locate the bucket containing the k-th element; update prefix/k; zero hists.
__global__ void ssi_scan(unsigned* __restrict__ ws, int pass) {
  __shared__ unsigned ssum[TPB];
  __shared__ unsigned sbucket, scum;
  int b = blockIdx.x, tid = threadIdx.x;
  int bins = (pass == 2) ? 1024 : 2048;
  int per  = bins >> 8;                      // bins per thread: 8 or 4

  for (int arr = 0; arr < 2; ++arr) {
    const unsigned* hist = ws + (arr == 0 ? WS_HIST_I : WS_HIST_T) + b * HBINS;
    unsigned* pref = ws + (arr == 0 ? WS_PREF_I : WS_PREF_T);
    unsigned* krem = ws + (arr == 0 ? WS_KREM_I : WS_KREM_T);
    unsigned k;
    if (pass == 0) {
      unsigned nvv = ws[WS_NVALID + b];
      k = nvv ? (nvv - 1u) >> 1 : 0u;        // torch lower-median index
    } else {
      k = krem[b];
    }
    if (tid == 0) { sbucket = (unsigned)(bins - 1); scum = k; }  // fallback
    unsigned local[8]; unsigned lsum = 0u;
    for (int j = 0; j < per; ++j) { local[j] = hist[tid * per + j]; lsum += local[j]; }
    ssum[tid] = lsum; __syncthreads();
    for (int off = 1; off < TPB; off <<= 1) {          // Hillis-Steele scan
      unsigned v = (tid >= off) ? ssum[tid - off] : 0u;
      __syncthreads();
      ssum[tid] += v;
      __syncthreads();
    }
    unsigned incl = ssum[tid], excl = incl - lsum;
    if (excl <= k && k < incl) {                        // exactly one thread
      unsigned cum = excl;
      for (int j = 0; j < per; ++j) {
        if (k < cum + local[j]) { sbucket = (unsigned)(tid * per + j); scum = cum; break; }
        cum += local[j];
      }
    }
    __syncthreads();
    if (tid == 0) {
      unsigned bucket = sbucket, cum = scum, p;
      if (pass == 0)      p = bucket << 21;
      else if (pass == 1) p = pref[b] | (bucket << 10);
      else                p = pref[b] | bucket;
      pref[b] = p;
      krem[b] = k - cum;
      if (pass == 2)
        ((float*)ws)[(arr == 0 ? WS_MED_I : WS_MED_T) + b] = funmap(p);
    }
    __syncthreads();
  }
  for (int i = tid; i < HBINS; i += TPB) {              // reset for next pass
    ws[WS_HIST_I + b * HBINS + i] = 0u;
    ws[WS_HIST_T + b * HBINS + i] = 0u;
  }
}

// Per-block partials of  sum |x - med| * m  for input and target.
__global__ void ssi_scale(const float* __restrict__ xi, const float* __restrict__ xt,
                          const unsigned char* __restrict__ mask,
                          unsigned* __restrict__ ws, int n) {
  __shared__ float sred[TPB];
  int b = blockIdx.x >> 6, chunk = blockIdx.x & (NB - 1);
  const float* f = (const float*)ws;
  float mi = f[WS_MED_I + b], mt = f[WS_MED_T + b];
  const float4* pi4 = (const float4*)(xi + (size_t)b * n);
  const float4* pt4 = (const float4*)(xt + (size_t)b * n);
  const uchar4* pm4 = (const uchar4*)(mask + (size_t)b * n);
  int nv4 = n >> 2;

  float ai = 0.0f, at = 0.0f;
  for (int v = chunk * TPB + threadIdx.x; v < nv4; v += NB * TPB) {
    float4 vi = pi4[v]; float4 vt = pt4[v]; uchar4 m4 = pm4[v];
    if (m4.x) { ai += fabsf(vi.x - mi); at += fabsf(vt.x - mt); }
    if (m4.y) { ai += fabsf(vi.y - mi); at += fabsf(vt.y - mt); }
    if (m4.z) { ai += fabsf(vi.z - mi); at += fabsf(vt.z - mt); }
    if (m4.w) { ai += fabsf(vi.w - mi); at += fabsf(vt.w - mt); }
  }
  if (chunk == 0) {
    const float* pi = xi + (size_t)b * n;
    const float* pt = xt + (size_t)b * n;
    const unsigned char* pm = mask + (size_t)b * n;
    for (int e = (nv4 << 2) + threadIdx.x; e < n; e += TPB)
      if (pm[e]) { ai += fabsf(pi[e] - mi); at += fabsf(pt[e] - mt); }
  }
  float si = block_sum_f(ai, sred);
  float st = block_sum_f(at, sred);
  if (threadIdx.x == 0) {
    ((float*)ws)[WS_PART_SI + b * NB + chunk] = si;
    ((float*)ws)[WS_PART_ST + b * NB + chunk] = st;
  }
}

// One wave per batch: WMMA-reduce the 64 scale partials (exact f32).
__global__ void ssi_reduce_scales(unsigned* __restrict__ ws) {
  int b = blockIdx.x;
  float* f = (float*)ws;
  float si = wave_sum_mult64(f + WS_PART_SI + b * NB, NB);
  float st = wave_sum_mult64(f + WS_PART_ST + b * NB, NB);
  if ((threadIdx.x & 31) == 0) {
    f[WS_SCALE_I + b] = si;
    f[WS_SCALE_T + b] = st;
  }
}

// Per-block partials of  sum |si - tn| * m.
__global__ void ssi_loss(const float* __restrict__ xi, const float* __restrict__ xt,
                         const unsigned char* __restrict__ mask,
                         unsigned* __restrict__ ws, int n) {
  __shared__ float sred[TPB];
  int b = blockIdx.x >> 6, chunk = blockIdx.x & (NB - 1);
  const float* f = (const float*)ws;
  float mi = f[WS_MED_I + b], mt = f[WS_MED_T + b];
  float ms = (float)ws[WS_NVALID + b];
  float ri = ms / f[WS_SCALE_I + b];
  float rt = ms / f[WS_SCALE_T + b];
  float tmin = funmap(ws[WS_TMINU + b]);
  float tmax = funmap(ws[WS_TMAXU + b]);
  float inv_range = 100.0f / (tmax - tmin);
  const float4* pi4 = (const float4*)(xi + (size_t)b * n);
  const float4* pt4 = (const float4*)(xt + (size_t)b * n);
  const uchar4* pm4 = (const uchar4*)(mask + (size_t)b * n);
  int nv4 = n >> 2;

  float acc = 0.0f;
  for (int v = chunk * TPB + threadIdx.x; v < nv4; v += NB * TPB) {
    float4 vi = pi4[v]; float4 vt = pt4[v]; uchar4 m4 = pm4[v];
    if (m4.x) acc += fabsf((vi.x - mi) * ri - ((vt.x - mt) * rt - tmin) * inv_range);
    if (m4.y) acc += fabsf((vi.y - mi) * ri - ((vt.y - mt) * rt - tmin) * inv_range);
    if (m4.z) acc += fabsf((vi.z - mi) * ri - ((vt.z - mt) * rt - tmin) * inv_range);
    if (m4.w) acc += fabsf((vi.w - mi) * ri - ((vt.w - mt) * rt - tmin) * inv_range);
  }
  if (chunk == 0) {
    const float* pi = xi + (size_t)b * n;
    const float* pt = xt + (size_t)b * n;
    const unsigned char* pm = mask + (size_t)b * n;
    for (int e = (nv4 << 2) + threadIdx.x; e < n; e += TPB)
      if (pm[e]) acc += fabsf((pi[e] - mi) * ri - ((pt[e] - mt) * rt - tmin) * inv_range);
  }
  float s = block_sum_f(acc, sred);
  if (threadIdx.x == 0) ((float*)ws)[WS_PART_L + b * NB + chunk] = s;
}

// Single wave: WMMA-accumulate all 2048 loss partials; divide by sum(nvalid).
__global__ void ssi_final(unsigned* __restrict__ ws, float* __restrict__ out) {
  int lane = threadIdx.x & 31;
  float num = wave_sum_mult64((float*)ws + WS_PART_L, BATCHES * NB);
  float den = (float)ws[WS_NVALID + lane];  // BATCHES == 32 == wave width
  #pragma unroll
  for (int d = 1; d < 32; d <<= 1) den += __shfl_xor(den, d, 32);
  if (lane == 0) out[0] = num / den;
}

// ---------------------------------------------------------------------------
extern "C" void kernel_launch(void* const* d_in, const int* in_sizes, int n_in,
                              void* d_out, int out_size, void* d_ws, size_t ws_size,
                              hipStream_t stream) {
  const float*         xi   = (const float*)d_in[0];
  const float*         xt   = (const float*)d_in[1];
  const unsigned char* mask = (const unsigned char*)d_in[2];  // bool, 1 byte/elem
  // d_in[3] = interpolate flag: bilinear align_corners resize with equal
  // in/out shapes is the identity, so it is a no-op here.
  unsigned* ws  = (unsigned*)d_ws;   // needs ~550 KB of scratch
  float*    out = (float*)d_out;

  int n = in_sizes[0] / BATCHES;     // elements per batch (C*H*W)
  dim3 egrid(BATCHES * NB), eblk(TPB);

  ssi_init<<<64, TPB, 0, stream>>>(ws);
  ssi_stats<<<egrid, eblk, 0, stream>>>(xi, xt, mask, ws, n);
  ssi_scan<<<BATCHES, TPB, 0, stream>>>(ws, 0);
  ssi_hist<<<egrid, eblk, 0, stream>>>(xi, xt, mask, ws, n, 1);
  ssi_scan<<<BATCHES, TPB, 0, stream>>>(ws, 1);
  ssi_hist<<<egrid, eblk, 0, stream>>>(xi, xt, mask, ws, n, 2);
  ssi_scan<<<BATCHES, TPB, 0, stream>>>(ws, 2);
  ssi_scale<<<egrid, eblk, 0, stream>>>(xi, xt, mask, ws, n);
  ssi_reduce_scales<<<BATCHES, 32, 0, stream>>>(ws);
  ssi_loss<<<egrid, eblk, 0, stream>>>(xi, xt, mask, ws, n);
  ssi_final<<<1, 32, 0, stream>>>(ws, out);
}